// Decoder_52630529245284
// MI455X (gfx1250) — compile-verified
//
#include <hip/hip_runtime.h>
#include <hip/hip_bf16.h>
#include <math.h>

typedef __attribute__((ext_vector_type(2))) float v2f;
typedef __attribute__((ext_vector_type(8))) float v8f;

#define B_SZ   128
#define N_SZ   1000
#define D_SZ   512
#define H_SZ   8
#define HD_SZ  64
#define M_SZ   (B_SZ * N_SZ)      // 128000 GEMM rows
#define NC_SZ  1024               // gv (512) + logit_key (512) columns
#define WNODE_LD 1536             // W_node leading dim
#define WNODE_COL0 512            // skip unused glimpse_key columns

static __device__ __forceinline__ void wait_async0() {
#if __has_builtin(__builtin_amdgcn_s_wait_asynccnt)
  __builtin_amdgcn_s_wait_asynccnt(0);
#else
  asm volatile("s_wait_asynccnt 0x0" ::: "memory");
#endif
}

// ---------------------------------------------------------------------------
// K0a: graph_embed[b,d] = mean_n embeddings[b,n,d]
__global__ __launch_bounds__(256) void dec_mean_kernel(
    const float* __restrict__ E, float* __restrict__ gmean) {
  const int b = blockIdx.x;
  const int d = blockIdx.y * 256 + threadIdx.x;
  const float* p = E + (size_t)b * N_SZ * D_SZ + d;
  float s = 0.f;
  for (int n = 0; n < N_SZ; ++n) s += p[(size_t)n * D_SZ];
  gmean[b * D_SZ + d] = s * (1.0f / (float)N_SZ);
}

// ---------------------------------------------------------------------------
// K0b: query[b,d] = gmean[b,:]@W_fixed[:,d] + sctx[b,:]@W_step[:,d]
__global__ __launch_bounds__(256) void dec_query_kernel(
    const float* __restrict__ gmean, const float* __restrict__ sctx,
    const float* __restrict__ Wf, const float* __restrict__ Ws,
    float* __restrict__ q) {
  const int b = blockIdx.x;
  const int d = blockIdx.y * 256 + threadIdx.x;
  float s = 0.f;
  for (int k = 0; k < D_SZ; ++k)
    s += gmean[b * D_SZ + k] * Wf[(size_t)k * D_SZ + d];
  for (int k = 0; k < 2 * D_SZ; ++k)
    s += sctx[b * 2 * D_SZ + k] * Ws[(size_t)k * D_SZ + d];
  q[b * D_SZ + d] = s;
}

// ---------------------------------------------------------------------------
// K1: KV[m, 0:1024] = E[m, :] @ W_node[:, 512:1536]   (gv ‖ logit_key)
// fp32 WMMA 16x16x4. grid (1000, 8), block 256 (8 waves).
// Workgroup tile: 128 (M) x 128 (N). Wave (wm,wn): M-tiles {2wm,2wm+1},
// N-tiles wn*64 .. wn*64+63  -> 8 accumulators.
#define TK 32
#define LDA 36   // padded K-stride (dwords); even (b64-aligned), gcd(36,64)=4
__global__ __launch_bounds__(256) void dec_gemm_kv_kernel(
    const float* __restrict__ E, const float* __restrict__ Wn,
    float* __restrict__ KV) {
  __shared__ float Ash[128][LDA];     // E tile, row-major [M][K]
  __shared__ float BshT[128][LDA];    // W tile transposed [N][K]

  const int m0 = blockIdx.x * 128;
  const int n0 = blockIdx.y * 128;
  const int tid = threadIdx.x;
  const int wave = tid >> 5;
  const int lane = tid & 31;
  const int wm = wave & 3;              // M pair index
  const int wn = wave >> 2;             // N half (0/1)
  const int lrow = lane & 15;
  const int khalf = (lane >> 4) << 1;   // 0 for lanes 0-15, 2 for 16-31

  v8f acc[8] = {};

  for (int kt = 0; kt < D_SZ; kt += TK) {
    // --- A tile (128x32) via async global->LDS, 16B per op, 4 per thread ---
    #pragma unroll
    for (int i = 0; i < 4; ++i) {
      const int idx = tid + i * 256;            // 0..1023 chunks of 4 floats
      const int r = idx >> 3, c4 = (idx & 7) << 2;
      const unsigned int lds_a = (unsigned int)(uintptr_t)(&Ash[r][c4]);
      const float* gp = E + (size_t)(m0 + r) * D_SZ + kt + c4;
      asm volatile("global_load_async_to_lds_b128 %0, %1, off"
                   :: "v"(lds_a), "v"(gp) : "memory");
    }
    // --- B tile (32x128) -> transposed scatter into BshT[N][K] ---
    #pragma unroll
    for (int i = 0; i < 4; ++i) {
      const int idx = tid + i * 256;            // 0..1023
      const int r = idx >> 5, c4 = (idx & 31) << 2;   // r = k row, c4 = col
      const float4 v = *(const float4*)(Wn + (size_t)(kt + r) * WNODE_LD +
                                        WNODE_COL0 + n0 + c4);
      BshT[c4 + 0][r] = v.x; BshT[c4 + 1][r] = v.y;
      BshT[c4 + 2][r] = v.z; BshT[c4 + 3][r] = v.w;
    }
    wait_async0();
    __syncthreads();

    #pragma unroll
    for (int k = 0; k < TK; k += 4) {
      // contiguous K-pair fragments -> single ds_load_b64 each
      const v2f a0 = *(const v2f*)(&Ash[wm * 32 + lrow][k + khalf]);
      const v2f a1 = *(const v2f*)(&Ash[wm * 32 + 16 + lrow][k + khalf]);
      #pragma unroll
      for (int j = 0; j < 4; ++j) {
        const v2f bfrag =
            *(const v2f*)(&BshT[wn * 64 + j * 16 + lrow][k + khalf]);
        acc[j] = __builtin_amdgcn_wmma_f32_16x16x4_f32(
            false, a0, false, bfrag, (short)0, acc[j], false, false);
        acc[4 + j] = __builtin_amdgcn_wmma_f32_16x16x4_f32(
            false, a1, false, bfrag, (short)0, acc[4 + j], false, false);
      }
    }
    __syncthreads();
  }

  // C layout: VGPR i -> M = i + 8*(lane>=16), N = lane%16
  const int cbase = n0 + wn * 64 + lrow;
  #pragma unroll
  for (int t = 0; t < 2; ++t) {
    const int rbase = m0 + wm * 32 + t * 16 + ((lane >> 4) << 3);
    #pragma unroll
    for (int j = 0; j < 4; ++j)
      #pragma unroll
      for (int i = 0; i < 8; ++i)
        KV[(size_t)(rbase + i) * NC_SZ + cbase + j * 16] = acc[t * 4 + j][i];
  }
}

// ---------------------------------------------------------------------------
// K2: fused masked 8-head attention + glimpse + logits + log_softmax, 1 WG / b
__global__ __launch_bounds__(256) void dec_attn_kernel(
    const float* __restrict__ KV, const float* __restrict__ q_g,
    const unsigned char* __restrict__ mask, const float* __restrict__ Wo,
    float* __restrict__ out) {
  __shared__ float q_s[D_SZ];
  __shared__ float sc[H_SZ * 1024];     // scores -> exp weights
  __shared__ float heads_s[D_SZ];
  __shared__ float gl[D_SZ];
  __shared__ float linv[H_SZ];
  __shared__ float red[8];
  __shared__ float logit_s[1024];

  const int b = blockIdx.x;
  const int tid = threadIdx.x;
  const int wave = tid >> 5;
  const int lane = tid & 31;
  const size_t rowbase = (size_t)b * N_SZ * NC_SZ;

  q_s[tid]       = q_g[b * D_SZ + tid];
  q_s[tid + 256] = q_g[b * D_SZ + tid + 256];
  __syncthreads();

  // scores[h][n] = (q_h . gv[n]_h) / sqrt(hd), masked
  for (int task = tid; task < N_SZ * H_SZ; task += 256) {
    const int n = task >> 3, h = task & 7;
    const float* kvp = KV + rowbase + (size_t)n * NC_SZ + h * HD_SZ;
    const float* qp = q_s + h * HD_SZ;
    float s = 0.f;
    #pragma unroll 8
    for (int k = 0; k < HD_SZ; ++k) s += qp[k] * kvp[k];
    s *= 0.125f;                                  // 1/sqrt(64)
    if (!mask[b * N_SZ + n]) s = -INFINITY;
    sc[h * 1024 + n] = s;
  }
  __syncthreads();

  // per-head softmax, one wave per head
  {
    const int h = wave;
    float m = -INFINITY;
    for (int n = lane; n < N_SZ; n += 32) m = fmaxf(m, sc[h * 1024 + n]);
    #pragma unroll
    for (int off = 16; off > 0; off >>= 1) m = fmaxf(m, __shfl_xor(m, off, 32));
    float l = 0.f;
    for (int n = lane; n < N_SZ; n += 32) {
      const float e = __expf(sc[h * 1024 + n] - m);   // exp(-inf - m) = 0
      sc[h * 1024 + n] = e;
      l += e;
    }
    #pragma unroll
    for (int off = 16; off > 0; off >>= 1) l += __shfl_xor(l, off, 32);
    if (lane == 0) linv[h] = 1.0f / l;
  }
  __syncthreads();

  // heads[d] = sum_n attn[h(d)][n] * gv[n][d]
  for (int d = tid; d < D_SZ; d += 256) {
    const int h = d >> 6;
    const float* kvp = KV + rowbase + d;
    float a = 0.f;
    for (int n = 0; n < N_SZ; ++n) a += sc[h * 1024 + n] * kvp[(size_t)n * NC_SZ];
    heads_s[d] = a * linv[h];
  }
  __syncthreads();

  // glimpse = heads @ W_out
  for (int d = tid; d < D_SZ; d += 256) {
    float s = 0.f;
    for (int k = 0; k < D_SZ; ++k) s += heads_s[k] * Wo[(size_t)k * D_SZ + d];
    gl[d] = s;
  }
  __syncthreads();

  // logits[n] = tanh(glimpse . lk[n] / sqrt(D)) * 10, masked
  for (int n = tid; n < N_SZ; n += 256) {
    const float* lkp = KV + rowbase + (size_t)n * NC_SZ + D_SZ;
    float s = 0.f;
    #pragma unroll 8
    for (int k = 0; k < D_SZ; ++k) s += gl[k] * lkp[k];
    s = tanhf(s * 0.044194173824159216f) * 10.0f;   // 1/sqrt(512)
    if (!mask[b * N_SZ + n]) s = -INFINITY;
    logit_s[n] = s;
  }
  __syncthreads();

  // block log_softmax over n
  float m = -INFINITY;
  for (int n = tid; n < N_SZ; n += 256) m = fmaxf(m, logit_s[n]);
  #pragma unroll
  for (int off = 16; off > 0; off >>= 1) m = fmaxf(m, __shfl_xor(m, off, 32));
  if (lane == 0) red[wave] = m;
  __syncthreads();
  float M = -INFINITY;
  #pragma unroll
  for (int i = 0; i < 8; ++i) M = fmaxf(M, red[i]);
  __syncthreads();

  float l = 0.f;
  for (int n = tid; n < N_SZ; n += 256) l += __expf(logit_s[n] - M);
  #pragma unroll
  for (int off = 16; off > 0; off >>= 1) l += __shfl_xor(l, off, 32);
  if (lane == 0) red[wave] = l;
  __syncthreads();
  float L = 0.f;
  #pragma unroll
  for (int i = 0; i < 8; ++i) L += red[i];
  const float logZ = M + __logf(L);

  for (int n = tid; n < N_SZ; n += 256)
    out[b * N_SZ + n] = logit_s[n] - logZ;
}

// ---------------------------------------------------------------------------
extern "C" void kernel_launch(void* const* d_in, const int* in_sizes, int n_in,
                              void* d_out, int out_size, void* d_ws, size_t ws_size,
                              hipStream_t stream) {
  const float*         E    = (const float*)d_in[0];          // [128,1000,512]
  const float*         sctx = (const float*)d_in[1];          // [128,1,1024]
  const unsigned char* mask = (const unsigned char*)d_in[2];  // [128,1,1000] bool
  const float*         Wn   = (const float*)d_in[3];          // [512,1536]
  const float*         Wf   = (const float*)d_in[4];          // [512,512]
  const float*         Ws   = (const float*)d_in[5];          // [1024,512]
  const float*         Wo   = (const float*)d_in[6];          // [512,512]
  float*               out  = (float*)d_out;                  // [128,1,1000]

  float* kv    = (float*)d_ws;                                // [128000,1024]
  float* gmean = kv + (size_t)M_SZ * NC_SZ;                   // [128,512]
  float* query = gmean + B_SZ * D_SZ;                         // [128,512]

  dec_mean_kernel<<<dim3(B_SZ, 2), 256, 0, stream>>>(E, gmean);
  dec_query_kernel<<<dim3(B_SZ, 2), 256, 0, stream>>>(gmean, sctx, Wf, Ws, query);
  dec_gemm_kv_kernel<<<dim3(M_SZ / 128, NC_SZ / 128), 256, 0, stream>>>(E, Wn, kv);
  dec_attn_kernel<<<B_SZ, 256, 0, stream>>>(kv, query, mask, Wo, out);
}